// GraphConvolution_47459388621283
// MI455X (gfx1250) — compile-verified
//
#include <hip/hip_runtime.h>

// GraphConvolution on MI455X (gfx1250, wave32):
//   out = segment_sum(adj[:,None]*x[src], dst) @ W + bias
//
// Phase 1 (memory/atomic bound, ~1.8 GB traffic @ 23.3 TB/s):
//   zero d_out, then one wave per edge does coalesced float4 gathers of
//   x[src] and global_atomic_add_f32 into d_out row dst.
// Phase 2 (in-place GEMM, fp32 WMMA for exactness vs fp32 reference):
//   each block stages its 32 support rows into LDS, syncs, then 8 waves
//   each compute a 16x128 slab with v_wmma_f32_16x16x4_f32 and overwrite
//   the same 32 rows (row-wise independence makes in-place safe).

typedef __attribute__((ext_vector_type(2))) float v2f;
typedef __attribute__((ext_vector_type(8))) float v8f;

#define N_NODES 100000
#define D 512
#define N_EDGES 800000
#define LDS_STRIDE 516   // 512 + 4 pad floats: bank-conflict-free A reads, 16B aligned rows

__global__ void gcn_zero_kernel(float4* __restrict__ out) {
    const int i = blockIdx.x * blockDim.x + threadIdx.x;   // grid sized exactly
    out[i] = make_float4(0.f, 0.f, 0.f, 0.f);
}

// One wave (32 lanes) per edge: 4 float4 gathers per lane (512B coalesced per
// iteration), 16 f32 atomics per lane into the destination row.
__global__ void gcn_scatter_kernel(const float* __restrict__ x,
                                   const float* __restrict__ adj,
                                   const int*   __restrict__ src,
                                   const int*   __restrict__ dst,
                                   float*       __restrict__ out) {
    const int lane = threadIdx.x & 31;
    const int wave = threadIdx.x >> 5;
    const int e    = blockIdx.x * 8 + wave;          // 100000 blocks * 8 waves = 800000
    const int   s  = src[e];
    const int   d  = dst[e];
    const float w  = adj[e];
    const float4* __restrict__ xr = (const float4*)(x + (size_t)s * D);
    float* __restrict__ orow = out + (size_t)d * D;
#pragma unroll
    for (int i = 0; i < 4; ++i) {
        const int j = lane + i * 32;                 // float4 index within row
        const float4 v = xr[j];
        atomicAdd(orow + j * 4 + 0, w * v.x);
        atomicAdd(orow + j * 4 + 1, w * v.y);
        atomicAdd(orow + j * 4 + 2, w * v.z);
        atomicAdd(orow + j * 4 + 3, w * v.w);
    }
}

// In-place dense transform: out[32 rows] = out[32 rows] @ W + bias.
// Fragment layouts per CDNA5 ISA 7.12.2 (32-bit, 16x16x4):
//   A: lane l -> (M = l%16, K = 2*(l/16) + {0,1})   [2 VGPRs]
//   B: lane l -> (K = 2*(l/16) + {0,1}, N = l%16)   [2 VGPRs]
//   C/D: VGPR v -> (M = v + 8*(l/16), N = l%16)     [8 VGPRs]
__global__ __launch_bounds__(256)
void gcn_gemm_kernel(const float* __restrict__ W,
                     const float* __restrict__ bias,
                     float*       __restrict__ out) {
    __shared__ float sA[32 * LDS_STRIDE];

    const int tid   = threadIdx.x;
    const int block = blockIdx.x;                    // 3125 blocks * 32 rows = 100000
    float* base = out + (size_t)block * 32 * D;

    // Stage the 32x512 support tile into LDS (fully coalesced b128 loads).
#pragma unroll
    for (int it = 0; it < 16; ++it) {
        const int f4  = tid + it * 256;              // float4 index in tile
        const int f   = f4 * 4;
        const int row = f >> 9;                      // /512
        const int col = f & 511;
        const float4 v = ((const float4*)base)[f4];
        *(float4*)(&sA[row * LDS_STRIDE + col]) = v;
    }
    __syncthreads();

    const int lane  = tid & 31;
    const int wave  = tid >> 5;
    const int strip = wave >> 2;                     // which 16-row strip (0..1)
    const int cg    = wave & 3;                      // which 128-col group (0..3)
    const int hl    = lane >> 4;                     // half-wave (0..1)
    const int l16   = lane & 15;

    const float* aRow = &sA[(strip * 16 + l16) * LDS_STRIDE];
    const int n0 = cg * 128 + l16;

    v8f acc[8] = {};                                 // 16x128 slab = 8 C fragments

    for (int k = 0; k < 512; k += 4) {
        // A fragment: two consecutive K values -> one ds_load_b64 per lane.
        const float2 a2 = *(const float2*)(aRow + k + hl * 2);
        v2f A; A.x = a2.x; A.y = a2.y;
        const float* wp = W + (size_t)(k + hl * 2) * D + n0;
#pragma unroll
        for (int t = 0; t < 8; ++t) {
            v2f B;
            B.x = wp[t * 16];                        // row k + 2*hl
            B.y = wp[t * 16 + D];                    // row k + 2*hl + 1
            acc[t] = __builtin_amdgcn_wmma_f32_16x16x4_f32(
                false, A, false, B, (short)0, acc[t], false, false);
        }
    }

    // Epilogue: bias add + in-place store (rows were consumed into LDS above).
#pragma unroll
    for (int t = 0; t < 8; ++t) {
        const int n = cg * 128 + t * 16 + l16;
        const float b = bias[n];
        float* op = out + (size_t)(block * 32 + strip * 16 + hl * 8) * D + n;
#pragma unroll
        for (int v = 0; v < 8; ++v) {
            op[(size_t)v * D] = acc[t][v] + b;
        }
    }
}

extern "C" void kernel_launch(void* const* d_in, const int* in_sizes, int n_in,
                              void* d_out, int out_size, void* d_ws, size_t ws_size,
                              hipStream_t stream) {
    const float* x    = (const float*)d_in[0];
    const float* adj  = (const float*)d_in[1];
    const float* W    = (const float*)d_in[2];
    const float* bias = (const float*)d_in[3];
    const int*   src  = (const int*)d_in[4];
    const int*   dst  = (const int*)d_in[5];
    float* out = (float*)d_out;

    (void)in_sizes; (void)n_in; (void)out_size; (void)d_ws; (void)ws_size;

    // N_NODES*D/4 float4s / 256 threads = 50000 blocks (exact)
    gcn_zero_kernel<<<50000, 256, 0, stream>>>((float4*)out);
    // 800000 edges / 8 waves per block = 100000 blocks (exact)
    gcn_scatter_kernel<<<100000, 256, 0, stream>>>(x, adj, src, dst, out);
    // 100000 rows / 32 per block = 3125 blocks (exact)
    gcn_gemm_kernel<<<3125, 256, 0, stream>>>(W, bias, out);
}